// RFCN_head_75419625718485
// MI455X (gfx1250) — compile-verified
//
#include <hip/hip_runtime.h>
#include <hip/hip_bf16.h>

typedef __attribute__((ext_vector_type(2))) float v2f;
typedef __attribute__((ext_vector_type(8))) float v8f;

struct SelState {
  unsigned gmaxKey;   // orderable key of global max of cls_score
  int      numPos;    // count(label > 0)
  unsigned prefix;    // radix-select accumulated high bits
  unsigned kRemain;   // remaining k within current prefix group
  unsigned T;         // exact key of the k-th largest loss_c
  unsigned tieNeed;   // how many ties at T to take (lowest index first)
  unsigned numHard;   // k
  unsigned pad;
};

// monotone float -> uint key (ascending)
__device__ __forceinline__ unsigned fkey(float f) {
  unsigned u = __float_as_uint(f);
  return (u & 0x80000000u) ? ~u : (u | 0x80000000u);
}
__device__ __forceinline__ float fromKey(unsigned k) {
  unsigned u = (k & 0x80000000u) ? (k & 0x7FFFFFFFu) : ~k;
  return __uint_as_float(u);
}

__global__ void k_init(SelState* st, unsigned* hists, const int* numHard) {
  int t = threadIdx.x;
  for (int i = t; i < 1024; i += blockDim.x) hists[i] = 0u;
  if (t == 0) {
    st->gmaxKey = 0u;
    st->numPos  = 0;
    st->prefix  = 0u;
    st->kRemain = (unsigned)numHard[0];
    st->T = 0u; st->tieNeed = 0u;
    st->numHard = (unsigned)numHard[0];
  }
}

__global__ void __launch_bounds__(256) k_gmax(const float* __restrict__ x,
                                              long long M, SelState* st) {
  __shared__ unsigned sh[256];
  unsigned k = 0u;
  const long long gtid   = (long long)blockIdx.x * blockDim.x + threadIdx.x;
  const long long stride = (long long)gridDim.x * blockDim.x;
  const long long M4 = M >> 2;
  const float4* __restrict__ x4 = (const float4*)x;
  for (long long i = gtid; i < M4; i += stride) {
    float4 v = x4[i];
    unsigned k0 = fkey(v.x), k1 = fkey(v.y), k2 = fkey(v.z), k3 = fkey(v.w);
    unsigned ka = k0 > k1 ? k0 : k1;
    unsigned kb = k2 > k3 ? k2 : k3;
    unsigned kc = ka > kb ? ka : kb;
    k = kc > k ? kc : k;
  }
  for (long long i = (M4 << 2) + gtid; i < M; i += stride) {  // tail
    unsigned kk = fkey(x[i]);
    k = kk > k ? kk : k;
  }
  sh[threadIdx.x] = k;
  __syncthreads();
  for (int s = 128; s > 0; s >>= 1) {
    if (threadIdx.x < s) {
      unsigned a = sh[threadIdx.x], b = sh[threadIdx.x + s];
      sh[threadIdx.x] = (b > a) ? b : a;
    }
    __syncthreads();
  }
  if (threadIdx.x == 0) atomicMax(&st->gmaxKey, sh[0]);
}

// 16 rows per wave. Wave's 16 contiguous rows (16*C floats) are staged into LDS
// with global_load_async_to_lds_b128 (ASYNCcnt), then WMMA f32 16x16x4 with
// B = ones computes exact f32 row sums of exp(x - gmax) from LDS.
// Fused: gathered logit (from LDS), loss_c, numPos, smooth-L1 block partials.
#define TILE_STRIDE 1312   // >= 16*C + 8 for C <= 81 (tail-chunk over-read pad)

__global__ void __launch_bounds__(256) k_lossc(
    const float* __restrict__ cls, const int* __restrict__ labels,
    const float* __restrict__ bbox, const float* __restrict__ targ,
    int N, int C, SelState* st,
    float* __restrict__ lossC, float* __restrict__ sl1Part) {
  const int tid  = threadIdx.x;
  const int lane = tid & 31;
  const int wv   = tid >> 5;                      // 8 waves / block
  const int rowBase = (blockIdx.x * 8 + wv) * 16;
  const int r = lane & 15;
  const int myRow = rowBase + r;

  __shared__ float tile[8][TILE_STRIDE];
  __shared__ float red[256];

  const float gmax = fromKey(st->gmaxKey);
  const bool wavePath = (rowBase + 16 <= N) && (16 * C + 8 <= TILE_STRIDE); // wave-uniform
  const bool rowValid = (lane < 16) && (myRow < N);

  float srow = 0.f, rowsum = 0.f, gathered = 0.f;
  int lab = 0;

  if (wavePath) {
    // ---- async-copy 16*C contiguous floats (16B-aligned, 4*C chunks) to LDS ----
    const char* tileG = (const char*)(cls + (size_t)rowBase * C);
    const unsigned ldsBase = (unsigned)(uintptr_t)&tile[wv][0];
    const int nChunks = (16 * C * 4) >> 4;        // 16-byte chunks
    for (int i = lane; i < nChunks; i += 32) {
      unsigned lofs = ldsBase + ((unsigned)i << 4);
      const char* gp = tileG + ((size_t)i << 4);
      asm volatile("global_load_async_to_lds_b128 %0, %1, off"
                   :: "v"(lofs), "v"(gp) : "memory");
    }
    asm volatile("s_wait_asynccnt 0x0" ::: "memory");

    // A-matrix (32-bit 16x4): lanes 0-15 hold K=0,1 ; lanes 16-31 hold K=2,3
    const float* tw = &tile[wv][r * C];
    const int cBase = (lane < 16) ? 0 : 2;
    v8f acc = {0.f,0.f,0.f,0.f,0.f,0.f,0.f,0.f};
    v2f bOnes = {1.0f, 1.0f};

    int k0 = 0;
    for (; k0 + 4 <= C; k0 += 4) {                // full chunks: no guards
      int col = k0 + cBase;
      v2f a;
      a.x = __expf(tw[col] - gmax);
      a.y = __expf(tw[col + 1] - gmax);
      acc = __builtin_amdgcn_wmma_f32_16x16x4_f32(false, a, false, bOnes,
                                                  (short)0, acc, false, false);
    }
    if (k0 < C) {                                 // tail: value selects, no branches
      int col = k0 + cBase;
      float v0 = tw[col];                         // in-bounds of padded tile row
      float v1 = tw[col + 1];
      v2f a;
      a.x = (col     < C) ? __expf(v0 - gmax) : 0.f;
      a.y = (col + 1 < C) ? __expf(v1 - gmax) : 0.f;
      acc = __builtin_amdgcn_wmma_f32_16x16x4_f32(false, a, false, bOnes,
                                                  (short)0, acc, false, false);
    }
    // D layout: VGPR q -> row q (lanes 0-15) / row 8+q (lanes 16-31); cols equal
#pragma unroll
    for (int q = 0; q < 8; ++q) {
      float vlo = __shfl(acc[q], 0, 32);
      float vhi = __shfl(acc[q], 16, 32);
      if (lane == q)     rowsum = vlo;
      if (lane == q + 8) rowsum = vhi;
    }
    if (rowValid) {
      lab = labels[myRow];
      gathered = tw[lab];                         // gather from LDS
    }
  } else if (rowValid) {                          // partial wave / large-C fallback
    const float* xr = cls + (size_t)myRow * C;
    for (int j = 0; j < C; ++j) rowsum += __expf(xr[j] - gmax);
    lab = labels[myRow];
    gathered = xr[lab];
  }

  if (rowValid) {
    float lv = (lab > 0) ? 100.0f : (__logf(rowsum) + gmax - gathered);
    lossC[myRow] = lv;
    if (lab > 0) {
      float4 bv = ((const float4*)bbox)[myRow];   // 16B-aligned b128 loads
      float4 tv = ((const float4*)targ)[myRow];
      float d0 = bv.x - tv.x, d1 = bv.y - tv.y, d2 = bv.z - tv.z, d3 = bv.w - tv.w;
      float a0 = fabsf(d0), a1 = fabsf(d1), a2 = fabsf(d2), a3 = fabsf(d3);
      srow  = (a0 < 1.0f) ? 0.5f * d0 * d0 : a0 - 0.5f;
      srow += (a1 < 1.0f) ? 0.5f * d1 * d1 : a1 - 0.5f;
      srow += (a2 < 1.0f) ? 0.5f * d2 * d2 : a2 - 0.5f;
      srow += (a3 < 1.0f) ? 0.5f * d3 * d3 : a3 - 0.5f;
    }
  }
  unsigned long long bal = __ballot(rowValid && lab > 0);
  if (lane == 0 && bal) atomicAdd(&st->numPos, (int)__popcll(bal));

  red[tid] = srow;
  __syncthreads();
  for (int s = 128; s > 0; s >>= 1) {
    if (tid < s) red[tid] += red[tid + s];
    __syncthreads();
  }
  if (tid == 0) sl1Part[blockIdx.x] = red[0];
}

__global__ void k_hist(const float* __restrict__ lossC, int N, SelState* st,
                       unsigned* __restrict__ hist, int pass) {
  __shared__ unsigned sh[256];
  for (int i = threadIdx.x; i < 256; i += blockDim.x) sh[i] = 0u;
  __syncthreads();
  const unsigned pfx = st->prefix;
  const int shift = 24 - 8 * pass;
  int stride = gridDim.x * blockDim.x;
  for (int i = blockIdx.x * blockDim.x + threadIdx.x; i < N; i += stride) {
    unsigned k = fkey(lossC[i]);
    bool match = (pass == 0) || ((k >> (shift + 8)) == pfx);
    if (match) atomicAdd(&sh[(k >> shift) & 0xFFu], 1u);
  }
  __syncthreads();
  for (int i = threadIdx.x; i < 256; i += blockDim.x)
    if (sh[i]) atomicAdd(&hist[i], sh[i]);
}

__global__ void k_scan(SelState* st, const unsigned* __restrict__ hist, int pass) {
  __shared__ unsigned sh[256];
  sh[threadIdx.x] = hist[threadIdx.x];
  __syncthreads();
  if (threadIdx.x == 0) {
    unsigned kRem = st->kRemain;
    unsigned cum = 0; int b = 0;
    for (int i = 255; i >= 0; --i) {
      unsigned c = sh[i];
      if (cum + c >= kRem) { b = i; break; }
      cum += c;
    }
    st->prefix  = (st->prefix << 8) | (unsigned)b;
    st->kRemain = kRem - cum;
    if (pass == 3) { st->T = st->prefix; st->tieNeed = st->kRemain; }
  }
}

__global__ void k_tiecount(const float* __restrict__ lossC, int N, SelState* st,
                           unsigned* __restrict__ tieCnt) {
  __shared__ unsigned wCnt[32];
  int tid = threadIdx.x;
  int i = blockIdx.x * blockDim.x + tid;
  bool isTie = (i < N) && (fkey(lossC[i]) == st->T);
  unsigned long long bal = __ballot(isTie);
  int lane = tid & 31, wid = tid >> 5;
  if (lane == 0) wCnt[wid] = (unsigned)__popcll(bal);
  __syncthreads();
  if (tid == 0) {
    unsigned s = 0;
    for (int w = 0; w < (int)(blockDim.x >> 5); ++w) s += wCnt[w];
    tieCnt[blockIdx.x] = s;
  }
}

__global__ void k_tiescan(const unsigned* __restrict__ tieCnt,
                          unsigned* __restrict__ tiePre, int nb) {
  __shared__ unsigned sh[1024];
  int t = threadIdx.x;
  if (t < nb) sh[t] = tieCnt[t];
  __syncthreads();
  if (t == 0) {
    unsigned run = 0;
    for (int i = 0; i < nb; ++i) { unsigned c = sh[i]; sh[i] = run; run += c; }
  }
  __syncthreads();
  if (t < nb) tiePre[t] = sh[t];
}

__global__ void __launch_bounds__(1024) k_ce(
    const float* __restrict__ cls, const int* __restrict__ labels,
    const float* __restrict__ lossC, int N, int C, SelState* st,
    const unsigned* __restrict__ tiePre,
    float* __restrict__ partW, float* __restrict__ partWN) {
  __shared__ unsigned wCnt[32], wPre[32];
  __shared__ float rW[1024], rWN[1024];
  int tid = threadIdx.x;
  int lane = tid & 31, wid = tid >> 5;
  int row = blockIdx.x * blockDim.x + tid;
  const unsigned T = st->T, tieNeed = st->tieNeed;
  const float wpos = (float)st->numPos / (float)st->numHard;

  unsigned k = (row < N) ? fkey(lossC[row]) : 0u;
  bool isTie = (row < N) && (k == T);
  bool gt    = (row < N) && (k > T);

  unsigned long long bal = __ballot(isTie);
  int laneRank = (int)__popcll(bal & ((1ull << lane) - 1ull));
  if (lane == 0) wCnt[wid] = (unsigned)__popcll(bal);
  __syncthreads();
  if (tid == 0) {
    unsigned run = 0;
    for (int w = 0; w < 32; ++w) { wPre[w] = run; run += wCnt[w]; }
  }
  __syncthreads();
  unsigned tieRank = tiePre[blockIdx.x] + wPre[wid] + (unsigned)laneRank;
  bool sel = gt || (isTie && tieRank < tieNeed);

  float myW = 0.f, myWN = 0.f;
  if (sel) {
    const float* xr = cls + (size_t)row * C;
    __builtin_prefetch(xr, 0, 0);                 // global_prefetch_b8
    float m = -3.4e38f;
#pragma unroll 4
    for (int j = 0; j < C; ++j) m = fmaxf(m, xr[j]);
    float s = 0.f;
#pragma unroll 4
    for (int j = 0; j < C; ++j) s += __expf(xr[j] - m);
    int lab = labels[row];
    float nll = m + __logf(s) - xr[lab];
    myW  = (lab == 0) ? wpos : 1.0f;
    myWN = myW * nll;
  }
  rW[tid] = myW; rWN[tid] = myWN;
  __syncthreads();
  for (int s2 = (int)(blockDim.x >> 1); s2 > 0; s2 >>= 1) {
    if (tid < s2) { rW[tid] += rW[tid + s2]; rWN[tid] += rWN[tid + s2]; }
    __syncthreads();
  }
  if (tid == 0) { partW[blockIdx.x] = rW[0]; partWN[blockIdx.x] = rWN[0]; }
}

__global__ void k_final(const float* __restrict__ partW, const float* __restrict__ partWN,
                        int nbCE, const float* __restrict__ sl1Part, int nbSL1,
                        SelState* st, float* __restrict__ out) {
  __shared__ float sA[256], sB[256], sC[256];
  int t = threadIdx.x;
  float a = 0.f, b = 0.f, c = 0.f;
  for (int i = t; i < nbCE; i += 256) { a += partW[i]; b += partWN[i]; }
  for (int i = t; i < nbSL1; i += 256) c += sl1Part[i];
  sA[t] = a; sB[t] = b; sC[t] = c;
  __syncthreads();
  for (int s = 128; s > 0; s >>= 1) {
    if (t < s) { sA[t] += sA[t + s]; sB[t] += sB[t + s]; sC[t] += sC[t + s]; }
    __syncthreads();
  }
  if (t == 0) {
    float numPos = (float)st->numPos;
    out[0] = sB[0] / sA[0];                 // weighted-mean CE
    out[1] = sC[0] / (numPos * 4.0f);       // smooth L1 mean over num_pos*4
  }
}

extern "C" void kernel_launch(void* const* d_in, const int* in_sizes, int n_in,
                              void* d_out, int out_size, void* d_ws, size_t ws_size,
                              hipStream_t stream) {
  const float* cls     = (const float*)d_in[0];
  const int*   label   = (const int*)d_in[1];
  const float* bbox    = (const float*)d_in[2];
  const float* targ    = (const float*)d_in[3];
  const int*   numHard = (const int*)d_in[6];
  float* out = (float*)d_out;

  const int N = in_sizes[1];
  const int C = in_sizes[0] / N;

  char* ws = (char*)d_ws;
  SelState* st     = (SelState*)ws;                              // 64 B
  unsigned* hists  = (unsigned*)(ws + 64);                       // 4*256 u32
  unsigned* tieCnt = (unsigned*)(ws + 64 + 4096);                // up to 1024 u32
  unsigned* tiePre = (unsigned*)(ws + 64 + 4096 + 4096);         // up to 1024 u32
  float*    partW  = (float*)(ws + 64 + 4096 + 8192);            // up to 1024 f32
  float*    partWN = (float*)(ws + 64 + 4096 + 8192 + 4096);     // up to 1024 f32
  float*    sl1Part= (float*)(ws + 64 + 4096 + 8192 + 8192);     // nbLossc f32
  const int nbLossc = (N + 127) / 128;                           // 256 thr = 8 waves * 16 rows
  size_t sl1Bytes = ((size_t)nbLossc * 4 + 63) / 64 * 64;
  float*    lossC  = (float*)(ws + 64 + 4096 + 8192 + 8192 + sl1Bytes);

  k_init<<<1, 1024, 0, stream>>>(st, hists, numHard);

  long long M = (long long)N * C;
  k_gmax<<<2048, 256, 0, stream>>>(cls, M, st);

  k_lossc<<<nbLossc, 256, 0, stream>>>(cls, label, bbox, targ, N, C, st, lossC, sl1Part);

  for (int p = 0; p < 4; ++p) {
    k_hist<<<1024, 256, 0, stream>>>(lossC, N, st, hists + p * 256, p);
    k_scan<<<1, 256, 0, stream>>>(st, hists + p * 256, p);
  }

  const int nbTie = (N + 1023) / 1024;
  k_tiecount<<<nbTie, 1024, 0, stream>>>(lossC, N, st, tieCnt);
  k_tiescan<<<1, 1024, 0, stream>>>(tieCnt, tiePre, nbTie);
  k_ce<<<nbTie, 1024, 0, stream>>>(cls, label, lossC, N, C, st, tiePre, partW, partWN);

  k_final<<<1, 256, 0, stream>>>(partW, partWN, nbTie, sl1Part, nbLossc, st, out);
}